// MultiHeadAttention_2035814499100
// MI455X (gfx1250) — compile-verified
//
#include <hip/hip_runtime.h>

// MHA reference degenerates: einsum 'nhqk,nhvl->nhql' contracts BOTH k and v,
// and softmax rows sum to 1, so  out[b,s,:] = ((sum_s value[b,s,:]) @ Wv + S*bv) @ Wo + bo
// broadcast over s. Bandwidth-bound: ~75 MB total => ~3.2 us at 23.3 TB/s.
// The two 16x1024x1024 GEMMs run on V_WMMA_F32_16X16X4_F32 (full fp32 precision).

typedef __attribute__((ext_vector_type(2))) float v2f;
typedef __attribute__((ext_vector_type(8))) float v8f;

#define B_    4
#define S_    2048
#define E_    1024
#define MPAD  16
#define CH_   16                 // sequence chunks for 2-stage reduction
#define SC_   (S_ / CH_)         // 128 s-values per chunk

// ---------------------------------------------------------------------------
// Kernel 1a: partial[c][b][e] = sum over 128 s-values of value[b, c*128+i, e].
// 64K threads / 256 blocks: enough independent HBM streams to saturate 23.3 TB/s.
// Consecutive threads -> consecutive e (fully coalesced 4B lanes per row).
// ---------------------------------------------------------------------------
__global__ void colsum_stage1_kernel(const float* __restrict__ value,
                                     float* __restrict__ partial) {
    int t  = blockIdx.x * blockDim.x + threadIdx.x;   // 0 .. CH_*B_*E_-1
    int e  = t & (E_ - 1);
    int rc = t >> 10;          // 0..63
    int b  = rc & (B_ - 1);    // batch 0..3
    int c  = rc >> 2;          // chunk 0..15
    const float* p = value + (size_t)b * S_ * E_ + (size_t)c * SC_ * E_ + e;
    float s = 0.0f;
    #pragma unroll 8
    for (int i = 0; i < SC_; ++i) s += p[(size_t)i * E_];
    partial[t] = s;
}

// ---------------------------------------------------------------------------
// Kernel 1b: valsum_p[16][E]; rows 0..3 = sum of the 16 chunk partials,
// rows 4..15 = 0 (M padding for the 16x16 WMMA tiles).
// ---------------------------------------------------------------------------
__global__ void colsum_stage2_kernel(const float* __restrict__ partial,
                                     float* __restrict__ valsum_p) {
    int t   = blockIdx.x * blockDim.x + threadIdx.x;  // 0 .. MPAD*E_-1
    int row = t >> 10;
    int e   = t & (E_ - 1);
    float s = 0.0f;
    if (row < B_) {
        #pragma unroll
        for (int c = 0; c < CH_; ++c)
            s += partial[((c << 2) + row) * E_ + e];
    }
    valsum_p[t] = s;
}

// ---------------------------------------------------------------------------
// Kernel 2: Cout[16][N] = A[16][K] @ Bm[K][N] + bias_scale * bias[N]
// One wave per 16-wide N tile (64 waves). K marched in steps of 4 with
// V_WMMA_F32_16X16X4_F32, fp32 accumulate in 8 VGPRs.
//
// Fragment layouts (ISA 7.12.2, wave32):
//   A 16x4 f32 : lane l<16  -> float2 {A[l][k],     A[l][k+1]}
//                lane l>=16 -> float2 {A[l-16][k+2], A[l-16][k+3]}
//   B 4x16 f32 : lane l<16  -> {B[k][n0+l],     B[k+1][n0+l]}
//                lane l>=16 -> {B[k+2][n0+l-16], B[k+3][n0+l-16]}
//   C/D 16x16  : VGPR j: M = j + 8*(lane>=16), N = n0 + (lane&15)
// EXEC is all-ones (no divergence around the WMMA).
// ---------------------------------------------------------------------------
__global__ void gemm16_wmma_kernel(const float* __restrict__ A,
                                   const float* __restrict__ Bm,
                                   const float* __restrict__ bias,
                                   float bias_scale,
                                   float* __restrict__ Cout) {
    const int K = E_, N = E_;
    int wave = (blockIdx.x * blockDim.x + threadIdx.x) >> 5;  // 0..63
    int lane = threadIdx.x & 31;
    int half = lane >> 4;        // 0: K pair 0..1, 1: K pair 2..3
    int l16  = lane & 15;
    int n0   = wave << 4;
    int n    = n0 + l16;

    v8f acc = {};
    const float* arow = A + l16 * K + (half << 1);            // 8B aligned
    const float* bcol = Bm + (size_t)(half << 1) * N + n;

    for (int k = 0; k < K; k += 4) {
        v2f a = *(const v2f*)(arow + k);
        v2f b;
        b.x = bcol[(size_t)k * N];
        b.y = bcol[(size_t)k * N + N];
        acc = __builtin_amdgcn_wmma_f32_16x16x4_f32(
            /*neg_a=*/false, a, /*neg_b=*/false, b,
            /*c_mod=*/(short)0, acc, /*reuse_a=*/false, /*reuse_b=*/false);
    }

    float bb = bias_scale * bias[n];
    #pragma unroll
    for (int j = 0; j < 8; ++j) {
        int m = (half << 3) + j;
        Cout[m * N + n] = acc[j] + bb;
    }
}

// ---------------------------------------------------------------------------
// Kernel 3: out[b][s][:] = rowout[b][:]  (broadcast over s), float4 stores.
// Pure store-bandwidth: 33.5 MB. rowout (16 KB used) stays resident in L2.
// ---------------------------------------------------------------------------
__global__ void broadcast_kernel(const float* __restrict__ rowout,
                                 float* __restrict__ out) {
    int gid    = blockIdx.x * blockDim.x + threadIdx.x;   // B*S*E/4 threads
    size_t idx = (size_t)gid * 4;
    int b = (int)(idx >> 21);            // S*E = 2^21
    int e = (int)(idx & (E_ - 1));       // idx multiple of 4 -> e multiple of 4
    const float4 v = *(const float4*)(rowout + b * E_ + e);
    *(float4*)(out + idx) = v;
}

// ---------------------------------------------------------------------------
// Inputs (setup_inputs order):
//  0:query 1:key 2:value 3:Wq 4:bq 5:Wk 6:bk 7:Wv 8:bv 9:Wo 10:bo
// q/k path provably cancels (softmax row-sum == 1), so only value/Wv/bv/Wo/bo
// are touched.
// ---------------------------------------------------------------------------
extern "C" void kernel_launch(void* const* d_in, const int* in_sizes, int n_in,
                              void* d_out, int out_size, void* d_ws, size_t ws_size,
                              hipStream_t stream) {
    (void)in_sizes; (void)n_in; (void)out_size; (void)ws_size;

    const float* value = (const float*)d_in[2];
    const float* Wv    = (const float*)d_in[7];
    const float* bv    = (const float*)d_in[8];
    const float* Wo    = (const float*)d_in[9];
    const float* bo    = (const float*)d_in[10];
    float* out = (float*)d_out;

    float* partial  = (float*)d_ws;             // [16][4][1024] 256 KB
    float* valsum_p = partial  + CH_ * B_ * E_; // [16][1024]     64 KB
    float* vsum_p   = valsum_p + MPAD * E_;     // [16][1024]     64 KB
    float* rowout   = vsum_p   + MPAD * E_;     // [16][1024]     64 KB

    // 1a) chunked column sums of value  (streams 33.5 MB, 256 blocks)
    colsum_stage1_kernel<<<(CH_ * B_ * E_) / 256, 256, 0, stream>>>(value, partial);

    // 1b) fold 16 partials -> valsum_p (rows 4..15 zeroed)
    colsum_stage2_kernel<<<(MPAD * E_) / 256, 256, 0, stream>>>(partial, valsum_p);

    // 2) vsum_p = valsum_p @ Wv + S*bv   (WMMA f32)
    gemm16_wmma_kernel<<<8, 256, 0, stream>>>(valsum_p, Wv, bv, (float)S_, vsum_p);

    // 3) rowout = vsum_p @ Wo + bo       (WMMA f32)
    gemm16_wmma_kernel<<<8, 256, 0, stream>>>(vsum_p, Wo, bo, 1.0f, rowout);

    // 4) out[b,s,:] = rowout[b,:]
    broadcast_kernel<<<(B_ * S_ * E_ / 4) / 256, 256, 0, stream>>>(rowout, out);
}